// MockGNN_64158221467971
// MI455X (gfx1250) — compile-verified
//
#include <hip/hip_runtime.h>

// RGCN forward for MI455X (gfx1250, wave32).
// out = x@W_root + bias + sum_r (segment_mean_r(x)) @ W[r]
// Strategy: aggregation commutes with the per-relation linear map, so we
// scatter-sum raw features per relation (atomics into an L2-resident 51MB
// scratch), then run one fp32 WMMA GEMM per relation accumulating into out.

typedef float v2f __attribute__((ext_vector_type(2)));
typedef float v8f __attribute__((ext_vector_type(8)));

#define DIMF 128
#define NRELS 8
#define KPAIRS (DIMF / 2)   // 64 k-pairs
#define WPAD 144            // float2 stride per kp-row: 288 dwords % 64 = 32
                            // -> lanes 0-15 (kp) and 16-31 (kp+1) hit disjoint
                            //    bank halves: conflict-free ds_load_b64

__device__ __forceinline__ int imin(int a, int b) { return a < b ? a : b; }

// ---------------------------------------------------------------- zero scratch
__global__ __launch_bounds__(256) void rgcn_zero(float* __restrict__ p, int n) {
  int i = blockIdx.x * blockDim.x + threadIdx.x;
  int stride = gridDim.x * blockDim.x;
  for (; i < n; i += stride) p[i] = 0.0f;
}

// --------------------------------------------------- per-relation scatter-sum
// One wave32 per edge: 4 floats/lane covers DIM=128. Atomic f32 adds land in
// the 51MB S buffer which is L2-resident on MI455X (192MB L2).
__global__ __launch_bounds__(256) void rgcn_scatter(
    const float* __restrict__ x, const int* __restrict__ edge_index,
    const int* __restrict__ edge_type, int rel,
    float* __restrict__ S, float* __restrict__ cnt, int E) {
  int e = (blockIdx.x * blockDim.x + threadIdx.x) >> 5;
  int lane = threadIdx.x & 31;
  if (e >= E) return;
  if (edge_type[e] != rel) return;  // uniform per wave (one edge per wave)
  int src = edge_index[e];
  int dst = edge_index[E + e];
  const float4* xs = (const float4*)(x + (size_t)src * DIMF);
  float4 v = xs[lane];
  float* sd = S + (size_t)dst * DIMF + lane * 4;
  __hip_atomic_fetch_add(sd + 0, v.x, __ATOMIC_RELAXED, __HIP_MEMORY_SCOPE_AGENT);
  __hip_atomic_fetch_add(sd + 1, v.y, __ATOMIC_RELAXED, __HIP_MEMORY_SCOPE_AGENT);
  __hip_atomic_fetch_add(sd + 2, v.z, __ATOMIC_RELAXED, __HIP_MEMORY_SCOPE_AGENT);
  __hip_atomic_fetch_add(sd + 3, v.w, __ATOMIC_RELAXED, __HIP_MEMORY_SCOPE_AGENT);
  if (lane == 0)
    __hip_atomic_fetch_add(cnt + dst, 1.0f, __ATOMIC_RELAXED, __HIP_MEMORY_SCOPE_AGENT);
}

// -------------------------------------------------------- fp32 WMMA GEMM pass
// C[n,128] (+)= (A[n,128] * rowscale) @ Wm[128,128]
// Block: 256 threads = 8 waves; each wave owns a 16-row strip x 128 cols
// (8 tiles of 16x16, K=128 -> 32 x V_WMMA_F32_16X16X4_F32 per tile).
// Wm is staged in LDS in k-pair-packed form: sW2[kp][n] = {W[2kp][n],
// W[2kp+1][n]} so every B fragment is one aligned ds_load_b64 (no shuffles).
__global__ __launch_bounds__(256) void rgcn_gemm(
    const float* __restrict__ A, const float* __restrict__ cnt,
    const float* __restrict__ Wm, const float* __restrict__ bias,
    float* __restrict__ out, int nrows, int accumulate) {
  __shared__ v2f sW2[KPAIRS * WPAD];  // 72 KB of the 320 KB WGP LDS

  int tid = threadIdx.x;
  // Cooperative load of the 128x128 weight matrix into packed LDS layout.
  #pragma unroll
  for (int j = 0; j < (KPAIRS * DIMF) / 256; ++j) {
    int i = tid + j * 256;
    int kp = i >> 7;      // k-pair index
    int n  = i & 127;     // output column
    v2f w;
    w.x = Wm[(2 * kp) * DIMF + n];
    w.y = Wm[(2 * kp + 1) * DIMF + n];
    sW2[kp * WPAD + n] = w;
  }
  __syncthreads();

  int wave = tid >> 5;
  int lane = tid & 31;
  int row0 = (blockIdx.x * 8 + wave) * 16;  // 16-row strip per wave
  int m = lane & 15;                        // row within strip / col within tile
  int khalf = (lane < 16) ? 0 : 2;          // ISA 16x4 f32 A-fragment layout

  // Fold the per-row mean scale into the A fragment; scale=0 + clamped
  // address for tail rows keeps EXEC all-ones through every WMMA.
  int row = row0 + m;
  float scale;
  if (row < nrows) {
    scale = (cnt != nullptr) ? (1.0f / fmaxf(cnt[row], 1.0f)) : 1.0f;
  } else {
    scale = 0.0f;
    row = nrows - 1;
  }
  const float* arow = A + (size_t)row * DIMF + khalf;
  // Per-lane B base: kp offset (khalf/2) plus column m; tile/t and kb offsets
  // fold into the ds_load immediate.
  const v2f* bbase = sW2 + (khalf >> 1) * WPAD + m;

  v8f acc[8];
  #pragma unroll
  for (int t = 0; t < 8; ++t) {
    int colg = t * 16 + m;
    if (accumulate) {
      // read-modify-write: seed accumulators with current out tile
      #pragma unroll
      for (int v = 0; v < 8; ++v) {
        int rr = imin(row0 + ((lane < 16) ? v : 8 + v), nrows - 1);
        acc[t][v] = out[(size_t)rr * DIMF + colg];
      }
    } else {
      float bv = bias[colg];  // C/D layout: all 8 elems share column colg
      #pragma unroll
      for (int v = 0; v < 8; ++v) acc[t][v] = bv;
    }
  }

  // K loop: 32 steps of K=4, 8 WMMAs per step (one per 16-col tile).
  for (int kb = 0; kb < DIMF; kb += 4) {
    v2f a;
    a.x = arow[kb] * scale;
    a.y = arow[kb + 1] * scale;
    const v2f* bk = bbase + (kb >> 1) * WPAD;
    #pragma unroll
    for (int t = 0; t < 8; ++t) {
      v2f b = bk[t * 16];  // {W[kb+khalf][n], W[kb+khalf+1][n]} in one b64
      acc[t] = __builtin_amdgcn_wmma_f32_16x16x4_f32(
          false, a, false, b, (short)0, acc[t], false, false);
    }
  }

  // Store D: VGPR v -> row (row0 + v) [lanes 0-15] / (row0 + 8 + v) [16-31].
  #pragma unroll
  for (int t = 0; t < 8; ++t) {
    int colg = t * 16 + m;
    #pragma unroll
    for (int v = 0; v < 8; ++v) {
      int rr = row0 + ((lane < 16) ? v : 8 + v);
      if (rr < nrows) out[(size_t)rr * DIMF + colg] = acc[t][v];
    }
  }
}

extern "C" void kernel_launch(void* const* d_in, const int* in_sizes, int n_in,
                              void* d_out, int out_size, void* d_ws, size_t ws_size,
                              hipStream_t stream) {
  const float* x      = (const float*)d_in[0];  // [N,128]
  const float* W      = (const float*)d_in[1];  // [8,128,128]
  const float* W_root = (const float*)d_in[2];  // [128,128]
  const float* bias   = (const float*)d_in[3];  // [128]
  const int*   ei     = (const int*)d_in[4];    // [2,E] (src row 0, dst row 1)
  const int*   et     = (const int*)d_in[5];    // [E]

  int N = in_sizes[0] / DIMF;
  int E = in_sizes[5];
  float* out = (float*)d_out;

  float* S   = (float*)d_ws;            // [N,128] per-relation feature sums
  float* cnt = S + (size_t)N * DIMF;    // [N]     per-relation degree

  int gemm_grid    = (N + 127) / 128;           // 128 rows per block
  int scatter_grid = (E * 32 + 255) / 256;      // one wave per edge
  int zero_n       = N * DIMF + N;
  int zero_grid    = 4096;

  // Root transform + bias (writes every row of out).
  rgcn_gemm<<<gemm_grid, 256, 0, stream>>>(x, nullptr, W_root, bias, out, N, 0);

  // Per relation: zero scratch, scatter-sum features + counts, GEMM-accumulate.
  for (int r = 0; r < NRELS; ++r) {
    rgcn_zero<<<zero_grid, 256, 0, stream>>>(S, zero_n);
    rgcn_scatter<<<scatter_grid, 256, 0, stream>>>(x, ei, et, r, S, cnt, E);
    rgcn_gemm<<<gemm_grid, 256, 0, stream>>>(
        S, cnt, W + (size_t)r * DIMF * DIMF, nullptr, out, N, 1);
  }
}